// ReshapePagedCache_58050777972854
// MI455X (gfx1250) — compile-verified
//
#include <hip/hip_runtime.h>

// Problem constants (match reference)
#define TOKENS   8192
#define HEADS    32
#define HEAD_DIM 128
#define NBLOCKS  2048
#define BLK_SZ   16
#define TOTAL_SLOTS (NBLOCKS * BLK_SZ)   // 32768

static constexpr size_t CACHE_ELEMS = (size_t)NBLOCKS * HEADS * BLK_SZ * HEAD_DIM; // 134,217,728 floats (512 MiB)
static constexpr size_t CACHE_F4    = CACHE_ELEMS / 4;                             // 33,554,432 float4
static constexpr size_t TOTAL_ROWS  = 2 * CACHE_ELEMS / HEAD_DIM;                  // 2,097,152 rows of 128 floats

// ---- gfx1250 async global<->LDS DMA (ASYNCcnt) path, probe-guarded ----------
#if defined(__has_builtin)
#if __has_builtin(__builtin_amdgcn_global_load_async_to_lds_b128) && \
    __has_builtin(__builtin_amdgcn_global_store_async_from_lds_b128)
#define HAVE_ASYNC_LDS 1
#endif
#endif
#ifndef HAVE_ASYNC_LDS
#define HAVE_ASYNC_LDS 0
#endif

// Builtin parameter types (confirmed by hipcc): v4i32 pointers,
// global side addrspace(1), LDS side addrspace(3).
typedef int v4i __attribute__((ext_vector_type(4)));
typedef __attribute__((address_space(1))) v4i GV4;
typedef __attribute__((address_space(3))) v4i LV4;

__device__ __forceinline__ void wait_async0() {
#if defined(__has_builtin)
#if __has_builtin(__builtin_amdgcn_s_wait_asynccnt)
  __builtin_amdgcn_s_wait_asynccnt(0);
#else
  asm volatile("s_wait_asynccnt 0" ::: "memory");
#endif
#else
  asm volatile("s_wait_asynccnt 0" ::: "memory");
#endif
}

// ============================ Fused single-pass path =========================

__global__ __launch_bounds__(256) void init_inverse_kernel(int* __restrict__ inv) {
  inv[blockIdx.x * 256u + threadIdx.x] = -1;
}

__global__ __launch_bounds__(256) void fill_inverse_kernel(
    const int* __restrict__ slots, int* __restrict__ inv) {
  const unsigned t = blockIdx.x * 256u + threadIdx.x;  // token id, < TOKENS
  const int s = slots[t];
  if (s >= 0 && s < TOTAL_SLOTS) inv[s] = (int)t;      // unique slots: race-free
}

// One wave handles 4 consecutive output rows (4 x 512B contiguous).
// row -> off = row&15, h = (row>>4)&31, blk = (row>>9)&2047, which = row>>20.
// Source per row: k/v row if inverse[slot]>=0, else the old cache row.
#define ROWS_PER_WAVE 4
__global__ __launch_bounds__(256) void fused_write_kernel(
    const float* __restrict__ k, const float* __restrict__ v,
    const float* __restrict__ k_cache, const float* __restrict__ v_cache,
    const int* __restrict__ inv, float* __restrict__ out) {
  const unsigned lane = threadIdx.x & 31u;
  const unsigned wib  = threadIdx.x >> 5;                  // wave in block, 0..7
  const size_t wave = (size_t)blockIdx.x * 8u + wib;
  const size_t row0 = wave * ROWS_PER_WAVE;

#if HAVE_ASYNC_LDS
  __shared__ float4 stage[8][ROWS_PER_WAVE][32];           // 16 KB
#pragma unroll
  for (int r = 0; r < ROWS_PER_WAVE; ++r) {
    const size_t row   = row0 + r;
    const unsigned off   = (unsigned)(row & 15u);
    const unsigned h     = (unsigned)((row >> 4) & 31u);
    const unsigned blk   = (unsigned)((row >> 9) & 2047u);
    const unsigned which = (unsigned)(row >> 20);          // 0 = k, 1 = v
    const unsigned slot  = blk * (unsigned)BLK_SZ + off;
    const int tok = inv[slot];                             // wave-uniform per row
    const float4* src;
    if (tok >= 0) {
      src = (const float4*)((which ? v : k) + ((size_t)tok * HEADS + h) * HEAD_DIM);
    } else {
      src = (const float4*)((which ? v_cache : k_cache) +
                            (row & (CACHE_ELEMS / HEAD_DIM - 1)) * HEAD_DIM);
    }
    __builtin_amdgcn_global_load_async_to_lds_b128(
        (GV4*)(src + lane), (LV4*)&stage[wib][r][lane], 0, 0);
  }
  wait_async0();  // all 4 row loads landed in LDS
#pragma unroll
  for (int r = 0; r < ROWS_PER_WAVE; ++r) {
    const size_t row = row0 + r;
    float4* dst = (float4*)(out + row * HEAD_DIM);
    __builtin_amdgcn_global_store_async_from_lds_b128(
        (GV4*)(dst + lane), (LV4*)&stage[wib][r][lane], 0, 0);
  }
  wait_async0();
#else
#pragma unroll
  for (int r = 0; r < ROWS_PER_WAVE; ++r) {
    const size_t row   = row0 + r;
    const unsigned off   = (unsigned)(row & 15u);
    const unsigned h     = (unsigned)((row >> 4) & 31u);
    const unsigned blk   = (unsigned)((row >> 9) & 2047u);
    const unsigned which = (unsigned)(row >> 20);
    const unsigned slot  = blk * (unsigned)BLK_SZ + off;
    const int tok = inv[slot];
    const float4* src;
    if (tok >= 0) {
      src = (const float4*)((which ? v : k) + ((size_t)tok * HEADS + h) * HEAD_DIM);
    } else {
      src = (const float4*)((which ? v_cache : k_cache) +
                            (row & (CACHE_ELEMS / HEAD_DIM - 1)) * HEAD_DIM);
    }
    float4 tmp = src[lane];
    ((float4*)(out + row * HEAD_DIM))[lane] = tmp;
  }
#endif
}

// ===================== Fallback two-pass path (no workspace) =================

#define CP_UNROLL 4
__global__ __launch_bounds__(256) void bulk_copy_kernel(
    const float* __restrict__ k_cache, const float* __restrict__ v_cache,
    float* __restrict__ out) {
  const float4* __restrict__ src =
      (const float4*)(blockIdx.y == 0 ? k_cache : v_cache);
  float4* __restrict__ dst = (float4*)(out + (size_t)blockIdx.y * CACHE_ELEMS);
  const size_t stride = (size_t)gridDim.x * blockDim.x;
  const size_t i0 = (size_t)blockIdx.x * blockDim.x + threadIdx.x;

#if HAVE_ASYNC_LDS
  __shared__ float4 stage[256 * CP_UNROLL];  // 16 KB
  for (size_t i = i0; i < CACHE_F4; i += stride * CP_UNROLL) {
#pragma unroll
    for (int j = 0; j < CP_UNROLL; ++j) {
      size_t idx = i + (size_t)j * stride;
      if (idx < CACHE_F4)
        __builtin_amdgcn_global_load_async_to_lds_b128(
            (GV4*)(src + idx), (LV4*)&stage[threadIdx.x + 256u * j], 0, 0);
    }
    wait_async0();
#pragma unroll
    for (int j = 0; j < CP_UNROLL; ++j) {
      size_t idx = i + (size_t)j * stride;
      if (idx < CACHE_F4)
        __builtin_amdgcn_global_store_async_from_lds_b128(
            (GV4*)(dst + idx), (LV4*)&stage[threadIdx.x + 256u * j], 0, 0);
    }
    wait_async0();
  }
#else
  for (size_t i = i0; i < CACHE_F4; i += stride) {
    float4 tmp = src[i];
    dst[i] = tmp;
  }
#endif
}

__global__ __launch_bounds__(256) void scatter_kernel(
    const float* __restrict__ k, const float* __restrict__ v,
    const int* __restrict__ slots, float* __restrict__ out) {
  unsigned tid = blockIdx.x * blockDim.x + threadIdx.x;
  const unsigned f4i = tid & 31u;
  unsigned r = tid >> 5;
  const unsigned h = r & (HEADS - 1); r >>= 5;
  const unsigned t = r & (TOKENS - 1); r >>= 13;
  const unsigned which = r;

  const int slot = slots[t];
  if (slot < 0 || slot >= TOTAL_SLOTS) return;
  const unsigned blk = (unsigned)slot >> 4;
  const unsigned off = (unsigned)slot & (BLK_SZ - 1);

  const float4* __restrict__ src =
      (const float4*)((which ? v : k) + ((size_t)t * HEADS + h) * HEAD_DIM);
  float4* __restrict__ dst =
      (float4*)(out + (size_t)which * CACHE_ELEMS +
                ((((size_t)blk * HEADS + h) * BLK_SZ + off) * HEAD_DIM));
  dst[f4i] = src[f4i];
}

// =============================================================================

extern "C" void kernel_launch(void* const* d_in, const int* in_sizes, int n_in,
                              void* d_out, int out_size, void* d_ws, size_t ws_size,
                              hipStream_t stream) {
  const float* k            = (const float*)d_in[0];
  const float* v            = (const float*)d_in[1];
  const float* k_cache      = (const float*)d_in[2];
  const float* v_cache      = (const float*)d_in[3];
  const int*   slot_mapping = (const int*)d_in[4];
  float* out = (float*)d_out;

  if (d_ws != nullptr && ws_size >= (size_t)TOTAL_SLOTS * sizeof(int)) {
    // Fused single-pass: ~2.0 GiB traffic (1 GiB read + 1 GiB write).
    int* inv = (int*)d_ws;
    init_inverse_kernel<<<TOTAL_SLOTS / 256, 256, 0, stream>>>(inv);
    fill_inverse_kernel<<<TOKENS / 256, 256, 0, stream>>>(slot_mapping, inv);
    const unsigned blocks = (unsigned)(TOTAL_ROWS / (ROWS_PER_WAVE * 8));  // 65,536
    fused_write_kernel<<<blocks, 256, 0, stream>>>(k, v, k_cache, v_cache, inv, out);
  } else {
    // Fallback two-pass: full copy then scatter (~2.5 GiB traffic).
    dim3 cgrid(32768, 2, 1);
    bulk_copy_kernel<<<cgrid, 256, 0, stream>>>(k_cache, v_cache, out);
    const unsigned total = 2u * TOKENS * HEADS * 32u;
    scatter_kernel<<<total / 256u, 256, 0, stream>>>(k, v, slot_mapping, out);
  }
}